// MinValues_48756468744801
// MI455X (gfx1250) — compile-verified
//
#include <hip/hip_runtime.h>
#include <math.h>

// ---- tile geometry -------------------------------------------------------
#define TW     128            // output tile width per block
#define TH      32            // output tile height per block
#define HALO_W (TW + 2)       // 130 valid halo columns
#define LDS_H  (TH + 2)       // 34 halo rows
#define LDS_W  (TW + 4)       // 132: stride padding (keeps rows bank-shifted)
#define CELLS  (LDS_H * HALO_W) // 4420 halo cells to fill

typedef __attribute__((address_space(1))) int gas_int;   // global int
typedef __attribute__((address_space(3))) int las_int;   // LDS int
typedef float __attribute__((ext_vector_type(4))) v4f;   // native float4

// Async global -> LDS copy of one dword (per-lane, EXEC-masked), ASYNCcnt.
__device__ __forceinline__ void async_copy_f32(const float* gsrc, float* ldst) {
#if __has_builtin(__builtin_amdgcn_global_load_async_to_lds_b32)
    __builtin_amdgcn_global_load_async_to_lds_b32(
        (gas_int*)gsrc, (las_int*)ldst, /*offset=*/0, /*cpol=*/0);
#else
    unsigned lds_off = (unsigned)(unsigned long long)ldst; // low 32b of generic = LDS offset
    asm volatile("global_load_async_to_lds_b32 %0, %1, off"
                 :: "v"(lds_off), "v"(gsrc)
                 : "memory");
#endif
}

__device__ __forceinline__ void wait_async_zero() {
#if __has_builtin(__builtin_amdgcn_s_wait_asynccnt)
    __builtin_amdgcn_s_wait_asynccnt(0);
#else
    asm volatile("s_wait_asynccnt 0" ::: "memory");
#endif
}

__global__ __launch_bounds__(256)
void MinValues_48756468744801_kernel(const float* __restrict__ x,
                                     float* __restrict__ out) {
    __shared__ float tile[LDS_H][LDS_W];

    const int tid   = threadIdx.x;             // 0..255
    const int col0  = blockIdx.x * TW;         // tile origin in plane
    const int row0  = blockIdx.y * TH;
    const long long plane = blockIdx.z;        // fused N*C plane index

    const float* __restrict__ xp = x   + plane * (256LL * 256LL);
    float*       __restrict__ op = out + plane * (256LL * 256LL);

    const float INF = __builtin_inff();

    // ---- fill phase: async DMA the (TH+2)x(TW+2) halo tile into LDS ------
    for (int idx = tid; idx < CELLS; idx += 256) {
        const int r  = idx / HALO_W;
        const int c  = idx - r * HALO_W;
        const int gr = row0 + r - 1;
        const int gc = col0 + c - 1;
        if ((unsigned)gr < 256u && (unsigned)gc < 256u) {
            async_copy_f32(xp + gr * 256 + gc, &tile[r][c]);
        } else {
            tile[r][c] = INF;                  // +inf padding (ds_store)
        }
    }
    wait_async_zero();       // my wave's async writes landed in LDS
    __syncthreads();         // everyone's did

    // ---- compute phase: 8 surrounding taps, center excluded --------------
    const int lane4 = (tid & 31) * 4;          // 4 adjacent output columns
    const int rbase = tid >> 5;                // 8 row-phases over TH=32
    for (int r = rbase; r < TH; r += 8) {
        float o[4];
        const int rr = r + 1;
#pragma unroll
        for (int c = 0; c < 4; ++c) {
            const int cc = lane4 + c + 1;
            float up = fminf(fminf(tile[rr - 1][cc - 1], tile[rr - 1][cc]),
                             tile[rr - 1][cc + 1]);
            float dn = fminf(fminf(tile[rr + 1][cc - 1], tile[rr + 1][cc]),
                             tile[rr + 1][cc + 1]);
            float md = fminf(tile[rr][cc - 1], tile[rr][cc + 1]);
            o[c] = fminf(fminf(up, dn), md);
        }
        v4f v = { o[0], o[1], o[2], o[3] };
        // NT hint: 256 MB output stream should not thrash the 192 MB L2
        __builtin_nontemporal_store(v, (v4f*)(op + (row0 + r) * 256 + col0 + lane4));
    }
}

extern "C" void kernel_launch(void* const* d_in, const int* in_sizes, int n_in,
                              void* d_out, int out_size, void* d_ws, size_t ws_size,
                              hipStream_t stream) {
    (void)n_in; (void)out_size; (void)d_ws; (void)ws_size;
    const float* x = (const float*)d_in[0];
    float* out = (float*)d_out;

    const int planes = in_sizes[0] / (256 * 256);   // N*C = 1024
    dim3 grid(256 / TW, 256 / TH, planes);          // (2, 8, 1024)
    dim3 block(256);                                // 8 wave32s
    MinValues_48756468744801_kernel<<<grid, block, 0, stream>>>(x, out);
}